// LightGCN_30554397343960
// MI455X (gfx1250) — compile-verified
//
#include <hip/hip_runtime.h>
#include <hip/hip_bf16.h>
#include <stdint.h>

#define TILE        1024      // edges staged per chunk
#define SPMM_BLOCK  256       // 8 wave32
#define SPMM_GRID   640

// ---- CDNA5 async global->LDS helpers (ASYNCcnt path, CDNA5 ISA) ----

__device__ __forceinline__ unsigned lds_off_of(const void* p) {
    // low 32 bits of a flat shared-aperture address == wave-relative LDS byte offset
    return (unsigned)(uintptr_t)p;
}

__device__ __forceinline__ void async_ld_b128(unsigned lds_byte_off,
                                              const void* gbase,
                                              unsigned gbyte_off) {
    // VDST = LDS byte-offset VGPR, VADDR = 32-bit offset VGPR, SADDR = 64-bit base
    asm volatile("global_load_async_to_lds_b128 %0, %1, %2"
                 :: "v"(lds_byte_off), "v"(gbyte_off), "s"(gbase)
                 : "memory");
}

__device__ __forceinline__ void wait_async_le3() {
    asm volatile("s_wait_asynccnt 3" ::: "memory");
}
__device__ __forceinline__ void wait_async_0() {
    asm volatile("s_wait_asynccnt 0" ::: "memory");
}

// ---- zero fill (d_out / d_ws are poisoned by the harness) ----

__global__ void lgcn_zero_kernel(float4* __restrict__ p, int n4) {
    int i = blockIdx.x * blockDim.x + threadIdx.x;
    if (i < n4) p[i] = float4{0.f, 0.f, 0.f, 0.f};
}

// ---- edge-parallel SpMM: out[dst] += w * h[src], 16 lanes per edge ----

__global__ void __launch_bounds__(SPMM_BLOCK)
lgcn_spmm_kernel(const int*   __restrict__ esrc,
                 const int*   __restrict__ edst,
                 const float* __restrict__ ew,
                 const float* __restrict__ h,
                 float*       __restrict__ out,
                 int E, int nChunks) {
    __shared__ __align__(16) int   s_src[2][TILE];
    __shared__ __align__(16) int   s_dst[2][TILE];
    __shared__ __align__(16) float s_w  [2][TILE];

    const int t      = threadIdx.x;
    const int lane16 = t & 15;   // dim chunk: floats [lane16*4, lane16*4+4)
    const int eslot  = t >> 4;   // edge slot within a 16-edge round

    // Stage 4 consecutive edges per thread into buffer `buf` via 3 async b128s.
    // Clamp (not predicate) the tail so EVERY wave always issues exactly 3
    // async ops per staged chunk -> s_wait_asynccnt immediates stay exact.
    auto stage = [&](int cBase, int buf) {
        int eg = cBase + 4 * t;
        if (eg > E - 4) eg = E - 4;       // E % 4 == 0; duplicates masked in compute
        unsigned gb = (unsigned)eg * 4u;  // byte offset, fits 32 bits
        async_ld_b128(lds_off_of(&s_src[buf][4 * t]), esrc, gb);
        async_ld_b128(lds_off_of(&s_dst[buf][4 * t]), edst, gb);
        async_ld_b128(lds_off_of(&s_w  [buf][4 * t]), ew,   gb);
    };

    int c = blockIdx.x;
    if (c >= nChunks) return;

    int buf = 0;
    stage(c * TILE, buf);                       // prologue: stage first chunk

    for (; c < nChunks; c += gridDim.x) {
        const int cn = c + gridDim.x;
        const bool have_next = (cn < nChunks);
        if (have_next) {
            stage(cn * TILE, buf ^ 1);          // prefetch next chunk (other buffer)
            wait_async_le3();                   // retire current chunk's 3 loads
        } else {
            wait_async_0();
        }
        __syncthreads();                        // LDS of `buf` visible to all waves

        const int cBase = c * TILE;
        #pragma unroll 4
        for (int r = 0; r < TILE / 16; ++r) {   // 16 edges per round across block
            const int el = r * 16 + eslot;
            const int eg = cBase + el;
            if (eg < E) {
                const int   s  = s_src[buf][el];
                const int   d  = s_dst[buf][el];
                const float wv = s_w  [buf][el];
                // coalesced 256B row gather: 16 lanes x b128 (L2-resident, 192MB)
                const float4 v = *((const float4*)(h + (size_t)s * 64) + lane16);
                float* o = out + (size_t)d * 64 + (size_t)lane16 * 4;
                atomicAdd(o + 0, v.x * wv);     // no-return global_atomic_add_f32
                atomicAdd(o + 1, v.y * wv);
                atomicAdd(o + 2, v.z * wv);
                atomicAdd(o + 3, v.w * wv);
            }
        }
        __syncthreads();                        // done reading `buf` before reuse
        buf ^= 1;
    }
}

// ---- launch: x -> out -> ws -> out (3 LightGCN layers) ----

extern "C" void kernel_launch(void* const* d_in, const int* in_sizes, int n_in,
                              void* d_out, int out_size, void* d_ws, size_t ws_size,
                              hipStream_t stream) {
    const float* x    = (const float*)d_in[0];
    const int*   esrc = (const int*)  d_in[1];
    const int*   edst = (const int*)  d_in[2];
    const float* ew   = (const float*)d_in[3];
    float* out = (float*)d_out;
    float* ws  = (float*)d_ws;

    const int E  = in_sizes[1];          // 1,600,000
    const int nf = in_sizes[0];          // NUM_NODES * 64
    const int n4 = nf / 4;
    const int nChunks = (E + TILE - 1) / TILE;

    dim3 zblk(256), zgrd((n4 + 255) / 256);
    int grid = SPMM_GRID < nChunks ? SPMM_GRID : nChunks;

    // layer 1: x -> out
    lgcn_zero_kernel<<<zgrd, zblk, 0, stream>>>((float4*)out, n4);
    lgcn_spmm_kernel<<<grid, SPMM_BLOCK, 0, stream>>>(esrc, edst, ew, x, out, E, nChunks);
    // layer 2: out -> ws
    lgcn_zero_kernel<<<zgrd, zblk, 0, stream>>>((float4*)ws, n4);
    lgcn_spmm_kernel<<<grid, SPMM_BLOCK, 0, stream>>>(esrc, edst, ew, out, ws, E, nChunks);
    // layer 3: ws -> out
    lgcn_zero_kernel<<<zgrd, zblk, 0, stream>>>((float4*)out, n4);
    lgcn_spmm_kernel<<<grid, SPMM_BLOCK, 0, stream>>>(esrc, edst, ew, ws, out, E, nChunks);
}